// FourierAttention_8710193676989
// MI455X (gfx1250) — compile-verified
//
#include <hip/hip_runtime.h>
#include <math.h>

typedef __attribute__((ext_vector_type(2))) float v2f;
typedef __attribute__((ext_vector_type(8))) float v8f;
typedef __attribute__((ext_vector_type(4))) unsigned int u32x4;
typedef __attribute__((ext_vector_type(4))) int i32x4;
typedef __attribute__((ext_vector_type(8))) int i32x8;

#define BB 16
#define LL 1024
#define HH 8
#define EE 64
#define DD 512      // d_model = H*E
#define LF 513      // rfft bins
#define LFP 528     // padded to 33*16
#define NT 33       // 16-wide tiles over LFP
#define BHN (BB*HH)

#if __has_builtin(__builtin_amdgcn_tensor_load_to_lds)
#define HAVE_TDM 1
#else
#define HAVE_TDM 0
#endif

__device__ __forceinline__ float sigmoidf_(float x) {
  return 1.0f / (1.0f + __expf(-x));
}

// f32 WMMA: D(16x16) = A(16x4) * B(4x16) + C   (wave32)
__device__ __forceinline__ v8f wmma_f32(v2f a, v2f b, v8f c) {
  return __builtin_amdgcn_wmma_f32_16x16x4_f32(
      /*neg_a=*/false, a, /*neg_b=*/false, b,
      /*c_mod=*/(short)0, c, /*reuse_a=*/false, /*reuse_b=*/false);
}

// ---------------------------------------------------------------------------
// TDM: async 2D tile load global -> LDS (Tensor Data Mover, TENSORcnt).
// Descriptor per CDNA5 ISA. 2D tile: groups 2/3 zero.
// This toolchain uses the 6-arg builtin form.
// ---------------------------------------------------------------------------
#if HAVE_TDM
__device__ __forceinline__ void tdm_load_2d(float* ldsbuf, const float* gsrc,
                                            unsigned tensor_d0, unsigned tensor_d1,
                                            unsigned tile_d0, unsigned tile_d1,
                                            unsigned long long stride0_elems) {
  unsigned long long ga = (unsigned long long)(uintptr_t)gsrc;
  unsigned lds_off = (unsigned)(uintptr_t)ldsbuf;  // low 32 bits = LDS offset
  u32x4 g0;
  g0[0] = 0x1u;                                           // count=1, user D#
  g0[1] = lds_off;                                        // lds_addr
  g0[2] = (unsigned)(ga & 0xFFFFFFFFu);                   // global_addr[31:0]
  g0[3] = (unsigned)((ga >> 32) & 0x01FFFFFFu) | (2u << 30);  // [56:32] | type=2
  i32x8 g1;
  g1[0] = (int)(2u << 16);                                // wg_mask=0, data_size=4B
  g1[1] = (int)((tensor_d0 & 0xFFFFu) << 16);             // tensor_dim0[15:0] @ bit48
  g1[2] = (int)(((tensor_d0 >> 16) & 0xFFFFu) |
                ((tensor_d1 & 0xFFFFu) << 16));           // dim0 hi | dim1 lo
  g1[3] = (int)(((tensor_d1 >> 16) & 0xFFFFu) |
                ((tile_d0 & 0xFFFFu) << 16));             // dim1 hi | tile_dim0
  g1[4] = (int)(tile_d1 & 0xFFFFu);                       // tile_dim1, tile_dim2=0
  g1[5] = (int)(unsigned)(stride0_elems & 0xFFFFFFFFu);   // dim0_stride[31:0]
  g1[6] = (int)((stride0_elems >> 32) & 0xFFFFu);         // dim0_stride[47:32]
  g1[7] = 0;
  i32x4 z4 = {0, 0, 0, 0};
  i32x8 z8 = {0, 0, 0, 0, 0, 0, 0, 0};
  __builtin_amdgcn_tensor_load_to_lds(g0, g1, z4, z4, z8, 0);
}
#else
__device__ __forceinline__ void stage_tile(float* ldsbuf, const float* gsrc,
                                           int rows, int rowlen, int gstride,
                                           int lane) {
  for (int i = lane; i < rows * rowlen; i += 32) {
    int r = i / rowlen, c = i - r * rowlen;
    ldsbuf[i] = gsrc[(size_t)r * gstride + c];
  }
}
#endif

// ---------------------------------------------------------------------------
// Kernel 1: Conv1d(512 -> 1024, k=3, pad=1) over L + GLU.
// Output q2 stored channel-major [B][D][L] so FFT reads are contiguous.
// ---------------------------------------------------------------------------
__global__ __launch_bounds__(256)
void conv_glu_kernel(const float* __restrict__ q, const float* __restrict__ W,
                     const float* __restrict__ bias, float* __restrict__ q2) {
  __shared__ float xs[DD][18];
  const int b = blockIdx.z;
  const int lbase = blockIdx.y * 16;
  const int c0 = blockIdx.x * 64;
  const int tid = threadIdx.x;

  for (int idx = tid; idx < DD * 18; idx += 256) {
    int ic = idx / 18, ll = idx - ic * 18;
    int l = lbase + ll - 1;
    xs[ic][ll] = (l >= 0 && l < LL) ? q[((size_t)b * LL + l) * DD + ic] : 0.0f;
  }
  __syncthreads();

  const int c = c0 + (tid & 63);
  const int lq = tid >> 6;
  const float* Wa = W + (size_t)c * DD * 3;
  const float* Wg = W + (size_t)(c + DD) * DD * 3;
  for (int j = 0; j < 4; ++j) {
    const int ll = lq * 4 + j;
    float acc_a = bias[c];
    float acc_g = bias[c + DD];
    for (int ic = 0; ic < DD; ++ic) {
      float x0 = xs[ic][ll], x1 = xs[ic][ll + 1], x2 = xs[ic][ll + 2];
      const float* wa = Wa + ic * 3;
      const float* wg = Wg + ic * 3;
      acc_a = fmaf(wa[0], x0, fmaf(wa[1], x1, fmaf(wa[2], x2, acc_a)));
      acc_g = fmaf(wg[0], x0, fmaf(wg[1], x1, fmaf(wg[2], x2, acc_g)));
    }
    q2[((size_t)b * DD + c) * LL + (lbase + ll)] = acc_a * sigmoidf_(acc_g);
  }
}

// ---------------------------------------------------------------------------
// Radix-2 DIT FFT of length 1024 in LDS (input already bit-reversed).
// ---------------------------------------------------------------------------
__device__ void fft1024(float2* a, int tid, float sign) {
  for (int s = 1; s <= 10; ++s) {
    const int len = 1 << s;
    const int half = len >> 1;
    for (int t = tid; t < 512; t += 256) {
      const int pos = t & (half - 1);
      const int pair = t >> (s - 1);
      const int i0 = pair * len + pos;
      const int i1 = i0 + half;
      float ang = sign * 6.283185307179586f * (float)pos / (float)len;
      float sn, cs;
      __sincosf(ang, &sn, &cs);
      float2 u = a[i0];
      float2 w = a[i1];
      float tr = cs * w.x - sn * w.y;
      float ti = cs * w.y + sn * w.x;
      a[i0] = make_float2(u.x + tr, u.y + ti);
      a[i1] = make_float2(u.x - tr, u.y - ti);
    }
    __syncthreads();
  }
}

// ---------------------------------------------------------------------------
// Kernel 2: rfft (ortho) of q2 / k / v along the sequence dim.
// qf/kf planes are stored E-PAIR INTERLEAVED: elem (e,x) at
//   bh*E*LFP + ((e>>1)*LFP + x)*2 + (e&1)
// so WMMA fragments (K pair per lane) are single aligned 8-byte loads.
// vf plane stays plain [E][LFP] (its fragments want adjacent x).
// ---------------------------------------------------------------------------
__global__ __launch_bounds__(256)
void rfft_kernel(const float* __restrict__ q2, const float* __restrict__ kin,
                 const float* __restrict__ vin,
                 float* __restrict__ qfr, float* __restrict__ qfi,
                 float* __restrict__ kfr, float* __restrict__ kfi,
                 float* __restrict__ vfr, float* __restrict__ vfi) {
  __shared__ float2 a[LL];
  const int chan = blockIdx.x;  // b*D + c, c = h*E + e
  const int src = blockIdx.y;   // 0:q2  1:k  2:v
  const int b = chan >> 9;
  const int c = chan & 511;
  const int tid = threadIdx.x;

  for (int i = tid; i < LL; i += 256) {
    float x;
    if (src == 0) {
      x = q2[(size_t)chan * LL + i];
    } else {
      const float* p = (src == 1) ? kin : vin;
      x = p[((size_t)b * LL + i) * DD + c];
    }
    int r = (int)(__brev((unsigned)i) >> 22);
    a[r] = make_float2(x, 0.0f);
  }
  __syncthreads();
  fft1024(a, tid, -1.0f);

  if (src == 2) {
    for (int x = tid; x < LFP; x += 256) {
      float2 vv = (x < LF) ? a[x] : make_float2(0.f, 0.f);
      vfr[(size_t)chan * LFP + x] = vv.x * 0.03125f;
      vfi[(size_t)chan * LFP + x] = vv.y * 0.03125f;
    }
  } else {
    float* pr = (src == 0) ? qfr : kfr;
    float* pi = (src == 0) ? qfi : kfi;
    const int e = chan & 63;
    const size_t pb = (size_t)(chan >> 6) * (EE * LFP) +
                      (size_t)(e >> 1) * (2 * LFP) + (e & 1);
    for (int x = tid; x < LFP; x += 256) {
      float2 vv = (x < LF) ? a[x] : make_float2(0.f, 0.f);
      pr[pb + 2 * (size_t)x] = vv.x * 0.03125f;
      pi[pb + 2 * (size_t)x] = vv.y * 0.03125f;
    }
  }
}

// ---------------------------------------------------------------------------
// Pass A: per-row L1 sums of gated |s|, s = qf conj(kf)^T / 8.
// One wave per (bh, x-tile). kf tiles (32x32 f32 incl. e-pair interleave)
// are TDM-prefetched to LDS double buffers; WMMA consumes ds_load_b64 frags.
// ---------------------------------------------------------------------------
__global__ __launch_bounds__(32)
void rowsum_kernel(const float* __restrict__ qfr, const float* __restrict__ qfi,
                   const float* __restrict__ kfr, const float* __restrict__ kfi,
                   float* __restrict__ rsum) {
  __shared__ float kbr[2][32 * 32];
  __shared__ float kbi[2][32 * 32];
  const int xt = blockIdx.x, bh = blockIdx.y;
  const int lane = threadIdx.x;
  const int m = lane & 15;
  const int kh = lane >> 4;
  const int x0 = xt * 16;
  const size_t base = (size_t)bh * EE * LFP;

  // A fragments (qf, pair-interleaved): one b64 load per k-step.
  v2f aR[16], aI[16];
#pragma unroll
  for (int ks = 0; ks < 16; ++ks) {
    const size_t off = base + ((size_t)(ks * 2 + kh) * LFP + x0 + m) * 2;
    aR[ks] = *(const v2f*)(qfr + off);
    aI[ks] = *(const v2f*)(qfi + off);
  }

#if HAVE_TDM
  tdm_load_2d(kbr[0], kfr + base, 2 * LFP, 32, 32, 32, 2 * LFP);
  tdm_load_2d(kbi[0], kfi + base, 2 * LFP, 32, 32, 32, 2 * LFP);
#endif

  v8f rs = {0.f, 0.f, 0.f, 0.f, 0.f, 0.f, 0.f, 0.f};

  for (int yt = 0; yt < NT; ++yt) {
    const float* krt = kbr[yt & 1];
    const float* kit = kbi[yt & 1];
#if HAVE_TDM
    if (yt + 1 < NT) {
      const size_t nof = base + (size_t)(yt + 1) * 16 * 2;
      tdm_load_2d(kbr[(yt + 1) & 1], kfr + nof, 2 * LFP, 32, 32, 32, 2 * LFP);
      tdm_load_2d(kbi[(yt + 1) & 1], kfi + nof, 2 * LFP, 32, 32, 32, 2 * LFP);
      __builtin_amdgcn_s_wait_tensorcnt(2);  // current tile's 2 loads done
    } else {
      __builtin_amdgcn_s_wait_tensorcnt(0);
    }
    asm volatile("" ::: "memory");
#else
    stage_tile(kbr[yt & 1], kfr + base + (size_t)yt * 32, 32, 32, 2 * LFP, lane);
    stage_tile(kbi[yt & 1], kfi + base + (size_t)yt * 32, 32, 32, 2 * LFP, lane);
    __syncthreads();
#endif
    v8f accR = {0.f, 0.f, 0.f, 0.f, 0.f, 0.f, 0.f, 0.f};
    v8f accI = accR;
#pragma unroll
    for (int ks = 0; ks < 16; ++ks) {
      const int li = (ks * 2 + kh) * 32 + m * 2;
      v2f bR = *(const v2f*)(krt + li);
      v2f bI = *(const v2f*)(kit + li);
      v2f bIn = -bI;
      accR = wmma_f32(aR[ks], bR, accR);
      accR = wmma_f32(aI[ks], bI, accR);
      accI = wmma_f32(aI[ks], bR, accI);
      accI = wmma_f32(aR[ks], bIn, accI);
    }
#pragma unroll
    for (int r = 0; r < 8; ++r) {
      float sr = accR[r] * 0.125f;
      float si = accI[r] * 0.125f;
      float mag = sqrtf(sr * sr + si * si);
      rs[r] += mag * (1.0f + sigmoidf_(mag));
    }
#if !HAVE_TDM
    __syncthreads();
#endif
  }
#pragma unroll
  for (int r = 0; r < 8; ++r) {
    float vv = rs[r];
    vv += __shfl_xor(vv, 1, 32);
    vv += __shfl_xor(vv, 2, 32);
    vv += __shfl_xor(vv, 4, 32);
    vv += __shfl_xor(vv, 8, 32);
    if (m == 0) rsum[(size_t)bh * LFP + x0 + r + kh * 8] = vv;
  }
}

// ---------------------------------------------------------------------------
// Pass B: recompute s tile, sc = s*(1+sigmoid(|s|))/rowsum (== attn*e^{i phase}),
// then of += vf * sc^T via a second WMMA GEMM. kf AND vf tiles TDM-prefetched.
// ---------------------------------------------------------------------------
__global__ __launch_bounds__(32)
void apply_kernel(const float* __restrict__ qfr, const float* __restrict__ qfi,
                  const float* __restrict__ kfr, const float* __restrict__ kfi,
                  const float* __restrict__ vfr, const float* __restrict__ vfi,
                  const float* __restrict__ rsum,
                  float* __restrict__ ofr, float* __restrict__ ofi) {
  __shared__ float kbr[2][32 * 32];
  __shared__ float kbi[2][32 * 32];
  __shared__ float vbr[2][64 * 16];
  __shared__ float vbi[2][64 * 16];
  __shared__ float scr[256];  // sc^T tile [y_local][x_local]
  __shared__ float sci[256];
  const int xt = blockIdx.x, bh = blockIdx.y;
  const int lane = threadIdx.x;
  const int m = lane & 15;
  const int kh = lane >> 4;
  const int x0 = xt * 16;
  const size_t base = (size_t)bh * EE * LFP;

  v2f aR[16], aI[16];
#pragma unroll
  for (int ks = 0; ks < 16; ++ks) {
    const size_t off = base + ((size_t)(ks * 2 + kh) * LFP + x0 + m) * 2;
    aR[ks] = *(const v2f*)(qfr + off);
    aI[ks] = *(const v2f*)(qfi + off);
  }

  float invR[8];
#pragma unroll
  for (int r = 0; r < 8; ++r) {
    float rsv = rsum[(size_t)bh * LFP + x0 + r + kh * 8];
    invR[r] = 1.0f / fmaxf(rsv, 1e-12f);
  }

  v8f oAR[4], oAI[4];
#pragma unroll
  for (int mt = 0; mt < 4; ++mt) {
    oAR[mt] = (v8f){0.f, 0.f, 0.f, 0.f, 0.f, 0.f, 0.f, 0.f};
    oAI[mt] = oAR[mt];
  }

#if HAVE_TDM
  tdm_load_2d(kbr[0], kfr + base, 2 * LFP, 32, 32, 32, 2 * LFP);
  tdm_load_2d(kbi[0], kfi + base, 2 * LFP, 32, 32, 32, 2 * LFP);
  tdm_load_2d(vbr[0], vfr + base, LFP, 64, 16, 64, LFP);
  tdm_load_2d(vbi[0], vfi + base, LFP, 64, 16, 64, LFP);
#endif

  for (int yt = 0; yt < NT; ++yt) {
    const int buf = yt & 1;
    const float* krt = kbr[buf];
    const float* kit = kbi[buf];
    const float* vrt = vbr[buf];
    const float* vit = vbi[buf];
#if HAVE_TDM
    if (yt + 1 < NT) {
      const size_t kof = base + (size_t)(yt + 1) * 32;
      const size_t vof = base + (size_t)(yt + 1) * 16;
      tdm_load_2d(kbr[buf ^ 1], kfr + kof, 2 * LFP, 32, 32, 32, 2 * LFP);
      tdm_load_2d(kbi[buf ^ 1], kfi + kof, 2 * LFP, 32, 32, 32, 2 * LFP);
      tdm_load_2d(vbr[buf ^ 1], vfr + vof, LFP, 64, 16, 64, LFP);
      tdm_load_2d(vbi[buf ^ 1], vfi + vof, LFP, 64, 16, 64, LFP);
      __builtin_amdgcn_s_wait_tensorcnt(4);  // current tile's 4 loads done
    } else {
      __builtin_amdgcn_s_wait_tensorcnt(0);
    }
    asm volatile("" ::: "memory");
#else
    stage_tile(kbr[buf], kfr + base + (size_t)yt * 32, 32, 32, 2 * LFP, lane);
    stage_tile(kbi[buf], kfi + base + (size_t)yt * 32, 32, 32, 2 * LFP, lane);
    stage_tile(vbr[buf], vfr + base + (size_t)yt * 16, 64, 16, LFP, lane);
    stage_tile(vbi[buf], vfi + base + (size_t)yt * 16, 64, 16, LFP, lane);
    __syncthreads();
#endif

    v8f accR = {0.f, 0.f, 0.f, 0.f, 0.f, 0.f, 0.f, 0.f};
    v8f accI = accR;
#pragma unroll
    for (int ks = 0; ks < 16; ++ks) {
      const int li = (ks * 2 + kh) * 32 + m * 2;
      v2f bR = *(const v2f*)(krt + li);
      v2f bI = *(const v2f*)(kit + li);
      v2f bIn = -bI;
      accR = wmma_f32(aR[ks], bR, accR);
      accR = wmma_f32(aI[ks], bI, accR);
      accI = wmma_f32(aI[ks], bR, accI);
      accI = wmma_f32(aR[ks], bIn, accI);
    }
    // gate + normalize, stash sc^T in LDS (D layout: row = r+8*kh, col = m)
#pragma unroll
    for (int r = 0; r < 8; ++r) {
      float sr = accR[r] * 0.125f;
      float si = accI[r] * 0.125f;
      float mag = sqrtf(sr * sr + si * si);
      float w = (1.0f + sigmoidf_(mag)) * invR[r];
      const int idx = m * 16 + (r + 8 * kh);
      scr[idx] = sr * w;
      sci[idx] = si * w;
    }
    __syncthreads();
    // of[e, x] += sum_y vf[e, y] * sc[x, y]   (A = vf 16x4, B = sc^T 4x16)
#pragma unroll
    for (int mt = 0; mt < 4; ++mt) {
#pragma unroll
      for (int ky = 0; ky < 4; ++ky) {
        const int e = mt * 16 + m;
        const int yl = ky * 4 + 2 * kh;
        v2f avR = *(const v2f*)(vrt + e * 16 + yl);
        v2f avI = *(const v2f*)(vit + e * 16 + yl);
        v2f avIn = -avI;
        const int kb = ky * 4 + 2 * kh;
        v2f bsR; bsR.x = scr[kb * 16 + m]; bsR.y = scr[(kb + 1) * 16 + m];
        v2f bsI; bsI.x = sci[kb * 16 + m]; bsI.y = sci[(kb + 1) * 16 + m];
        oAR[mt] = wmma_f32(avR, bsR, oAR[mt]);
        oAR[mt] = wmma_f32(avIn, bsI, oAR[mt]);
        oAI[mt] = wmma_f32(avR, bsI, oAI[mt]);
        oAI[mt] = wmma_f32(avI, bsR, oAI[mt]);
      }
    }
    __syncthreads();
  }

#pragma unroll
  for (int mt = 0; mt < 4; ++mt) {
#pragma unroll
    for (int r = 0; r < 8; ++r) {
      const int e = mt * 16 + r + 8 * kh;
      const int x = x0 + m;
      ofr[base + (size_t)e * LFP + x] = oAR[mt][r];
      ofi[base + (size_t)e * LFP + x] = oAI[mt][r];
    }
  }
}

// ---------------------------------------------------------------------------
// Kernel 5: irfft (ortho) via hermitian extension + inverse complex FFT.
// ---------------------------------------------------------------------------
__global__ __launch_bounds__(256)
void irfft_kernel(const float* __restrict__ ofr, const float* __restrict__ ofi,
                  float* __restrict__ out) {
  __shared__ float2 a[LL];
  const int chan = blockIdx.x;  // b*D + c
  const int b = chan >> 9;
  const int c = chan & 511;
  const int tid = threadIdx.x;

  for (int x = tid; x < LL; x += 256) {
    float re, im;
    if (x <= 512) {
      re = ofr[(size_t)chan * LFP + x];
      im = ofi[(size_t)chan * LFP + x];
    } else {
      re = ofr[(size_t)chan * LFP + (LL - x)];
      im = -ofi[(size_t)chan * LFP + (LL - x)];
    }
    int r = (int)(__brev((unsigned)x) >> 22);
    a[r] = make_float2(re, im);
  }
  __syncthreads();
  fft1024(a, tid, 1.0f);
  for (int l = tid; l < LL; l += 256) {
    out[((size_t)b * LL + l) * DD + c] = a[l].x * 0.03125f;
  }
}

// ---------------------------------------------------------------------------
extern "C" void kernel_launch(void* const* d_in, const int* in_sizes, int n_in,
                              void* d_out, int out_size, void* d_ws, size_t ws_size,
                              hipStream_t stream) {
  (void)in_sizes; (void)n_in; (void)out_size; (void)ws_size;
  const float* q    = (const float*)d_in[0];
  const float* k    = (const float*)d_in[1];
  const float* v    = (const float*)d_in[2];
  const float* W    = (const float*)d_in[3];
  const float* bias = (const float*)d_in[4];
  float* out = (float*)d_out;

  float* ws = (float*)d_ws;
  const size_t Q2N = (size_t)BB * DD * LL;        // 8,388,608 floats
  const size_t PL  = (size_t)BB * HH * EE * LFP;  // 4,325,376 floats per plane
  float* q2  = ws;
  float* qfr = q2 + Q2N;
  float* qfi = qfr + PL;
  float* kfr = qfi + PL;
  float* kfi = kfr + PL;
  float* vfr = kfi + PL;
  float* vfi = vfr + PL;
  float* ofr = vfi + PL;
  float* ofi = ofr + PL;
  float* rsum = ofi + PL;  // BHN * LFP floats

  conv_glu_kernel<<<dim3(DD / 64, LL / 16, BB), 256, 0, stream>>>(q, W, bias, q2);
  rfft_kernel<<<dim3(BB * DD, 3), 256, 0, stream>>>(q2, k, v, qfr, qfi, kfr, kfi, vfr, vfi);
  rowsum_kernel<<<dim3(NT, BHN), 32, 0, stream>>>(qfr, qfi, kfr, kfi, rsum);
  apply_kernel<<<dim3(NT, BHN), 32, 0, stream>>>(qfr, qfi, kfr, kfi, vfr, vfi, rsum, ofr, ofi);
  irfft_kernel<<<dim3(BB * DD), 256, 0, stream>>>(ofr, ofi, out);
}